// MeshUnpool_89489938579754
// MI455X (gfx1250) — compile-verified
//
#include <hip/hip_runtime.h>
#include <stdint.h>

// Problem constants (match reference)
#define Bc 8
#define NS 12288   // N_SMALL
#define NF 24576   // N_FULL
#define Cc 256     // channels
#define Kc 4096    // chain length

typedef float v4f __attribute__((ext_vector_type(4)));

// ---------------------------------------------------------------------------
// Kernel 1: per-batch index-chain resolution.
//   src[r] starts as r; for each step s (reverse column order): src[t]=src[f].
//   Then gidx[b,r] = position of src[r] in sorted mask_idx[b], or -1.
// One block per batch. All index state lives in LDS (88 KB < 320 KB/WGP).
// ---------------------------------------------------------------------------
__global__ __launch_bounds__(256) void mesh_unpool_resolve(
    const long long* __restrict__ mask_idx,  // [B, NS] int64, sorted unique
    const long long* __restrict__ order,     // [B, 2, K] int64
    int* __restrict__ gidx)                  // [B, NF] out: image row or -1
{
    __shared__ unsigned short src[NF];   // 48 KB
    __shared__ unsigned short msk[NS];   // 24 KB
    __shared__ unsigned int   ord[Kc];   // 16 KB, packed (f | t<<16), reversed

    const int b   = blockIdx.x;
    const int tid = threadIdx.x;

    // Identity init, two u16 indices packed per b32 DS store.
    {
        unsigned int* src32 = (unsigned int*)src;
        for (int i = tid; i < NF / 2; i += 256) {
            const unsigned r = (unsigned)(2 * i);
            src32[i] = r | ((r + 1u) << 16);
        }
    }
    for (int i = tid; i < NS; i += 256)
        msk[i] = (unsigned short)mask_idx[(size_t)b * NS + i];

    const long long* ob = order + (size_t)b * 2 * Kc;
    for (int s = tid; s < Kc; s += 256) {
        const int c = Kc - 1 - s;                    // reverse column order
        const unsigned f = (unsigned)ob[c];          // row 0 = source
        const unsigned t = (unsigned)ob[Kc + c];     // row 1 = target
        ord[s] = f | (t << 16);
    }
    __syncthreads();

    // Chain resolution on wave 0: speculative 32-wide chunks.
    if (tid < 32) {
        const int lane = tid;
        for (int base = 0; base < Kc; base += 32) {
            const unsigned u = ord[base + lane];
            const unsigned f = u & 0xffffu;
            const unsigned t = u >> 16;
            const unsigned short val = src[f];   // pre-chunk state

            // RAW hazard: an earlier lane in this chunk writes my source.
            // WAW: if a later lane writes the same target, it wins.
            bool hazard = false;
            bool last   = true;
            #pragma unroll
            for (int i = 0; i < 32; ++i) {
                const unsigned ti = (unsigned)__shfl((int)t, i, 32);
                if (i < lane && ti == f) hazard = true;
                if (i > lane && ti == t) last = false;
            }
            if (__any(hazard)) {
                // Rare (~2%): replay this chunk serially, exact semantics.
                if (lane == 0) {
                    for (int i = 0; i < 32; ++i) {
                        const unsigned w = ord[base + i];
                        src[w >> 16] = src[w & 0xffffu];
                    }
                }
            } else if (last) {
                src[t] = val;
            }
        }
    }
    __syncthreads();

    // pos(j): binary search over sorted mask in LDS; -1 if absent (zero row).
    for (int r = tid; r < NF; r += 256) {
        const unsigned j = src[r];
        int lo = 0, hi = NS - 1, g = -1;
        while (lo <= hi) {
            const int mid = (lo + hi) >> 1;
            const unsigned v = msk[mid];
            if (v == j) { g = mid; break; }
            if (v < j) lo = mid + 1; else hi = mid - 1;
        }
        gidx[(size_t)b * NF + r] = g;
    }
}

// ---------------------------------------------------------------------------
// Kernel 2: bandwidth kernel. One wave32 per output row; 2x float4 per lane
// (row = 256 floats = 64 x b128). Nontemporal stores for the 192 MB stream;
// global_prefetch_b8 on the gathered source row. Gather index forced scalar
// so the zero-row branch is a wave-level s_cbranch, not EXEC masking.
// ---------------------------------------------------------------------------
__global__ __launch_bounds__(256) void mesh_unpool_copy(
    const float* __restrict__ images,  // [B, NS, C]
    const int*   __restrict__ gidx,    // [B*NF]
    float*       __restrict__ out)     // [B, NF, C]
{
    const int row  = blockIdx.x * 8 + (threadIdx.x >> 5);  // 8 waves/block
    const int lane = threadIdx.x & 31;
    const int b    = row / NF;

    // row (hence g) is wave-uniform: pin it to an SGPR.
    const int g = __builtin_amdgcn_readfirstlane(gidx[row]);

    v4f v0 = {0.f, 0.f, 0.f, 0.f};
    v4f v1 = {0.f, 0.f, 0.f, 0.f};
    if (g >= 0) {
        const v4f* s = (const v4f*)(images + ((size_t)b * NS + g) * Cc);
        __builtin_prefetch(s + lane, 0, 0);   // gfx1250 global_prefetch_b8
        v0 = s[lane];
        v1 = s[lane + 32];
    }
    v4f* d = (v4f*)(out + (size_t)row * Cc);
    __builtin_nontemporal_store(v0, d + lane);
    __builtin_nontemporal_store(v1, d + lane + 32);
}

// ---------------------------------------------------------------------------
extern "C" void kernel_launch(void* const* d_in, const int* in_sizes, int n_in,
                              void* d_out, int out_size, void* d_ws, size_t ws_size,
                              hipStream_t stream) {
    const float*     images   = (const float*)d_in[0];      // [B, NS, C] f32
    const long long* mask_idx = (const long long*)d_in[1];  // [B, NS] i64
    const long long* order    = (const long long*)d_in[2];  // [B, 2, K] i64

    int* gidx = (int*)d_ws;  // B*NF ints = 768 KB scratch, fully rewritten

    hipLaunchKernelGGL(mesh_unpool_resolve, dim3(Bc), dim3(256), 0, stream,
                       mask_idx, order, gidx);

    hipLaunchKernelGGL(mesh_unpool_copy, dim3((Bc * NF) / 8), dim3(256), 0, stream,
                       images, gidx, (float*)d_out);
}